// Model_66297115181573
// MI455X (gfx1250) — compile-verified
//
#include <hip/hip_runtime.h>
#include <math.h>

// ---- constants mirroring the reference ----
#define N_F          1.0e6f                 // N = ND * NINFLATE
#define INV_PRIOR_N  (1.0f / N_F)           // 1/(PRIOR_SIG^2 * N), PRIOR_SIG = 1
#define ALPHA_C      0.05f
#define BETA1_C      0.9f
#define BETA2_C      0.999f
#define EPS_C        1e-8f
#define LR_C         1e-4f
// TEMPERATURE == 1.0 and ND_NOISE == 1.0 -> folded out

typedef float v4f __attribute__((ext_vector_type(4)));

struct Upd { float mom, m, v; };

__device__ __forceinline__ Upd update_one(float p, float p0, float g, float m,
                                          float v, float mom, float nz,
                                          float bc1, float bc2) {
    // grad of potential energy U (TEMPERATURE = 1)
    float grad_U = g + (p - p0) * INV_PRIOR_N;
    // Adam EMAs
    float m_new = BETA1_C * m + (1.0f - BETA1_C) * grad_U;
    float v_new = BETA2_C * v + (1.0f - BETA2_C) * (grad_U * grad_U);
    // bias correction
    float m_hat = m_new * bc1;
    float v_hat = v_new * bc2;
    float denom     = sqrtf(v_hat) + EPS_C;
    float inv_denom = 1.0f / denom;                 // precond_term
    float precond_grad = m_hat * inv_denom;
    float noise_scale  = sqrtf((2.0f * ALPHA_C / N_F) * inv_denom);
    float mom_new = mom * (1.0f - ALPHA_C) + LR_C * precond_grad + noise_scale * nz;
    Upd r; r.mom = mom_new; r.m = m_new; r.v = v_new;
    return r;
}

__global__ __launch_bounds__(256) void sghmc_adam_fused(
    const float* __restrict__ p,   const float* __restrict__ p0,
    const float* __restrict__ g,   const float* __restrict__ m,
    const float* __restrict__ v,   const float* __restrict__ mom,
    const float* __restrict__ nz,  const int*   __restrict__ t_ptr,
    float* __restrict__ out, unsigned int n)
{
    // 32-bit element index: all byte offsets (max 3*P*4 ~ 402 MB) fit in u32,
    // enabling SADDR(64b SGPR base) + 32-bit VGPR offset global addressing.
    const unsigned int base =
        (blockIdx.x * blockDim.x + threadIdx.x) * 4u;
    if (base >= n) return;

    // uniform bias-correction scalars (t is a runtime device scalar;
    // compiler scalarizes this entire block onto the SALU)
    const float tf  = (float)t_ptr[0];
    const float bc1 = 1.0f / (1.0f - powf(BETA1_C, tf));
    const float bc2 = 1.0f / (1.0f - powf(BETA2_C, tf));

    float* __restrict__ out_mom = out;
    float* __restrict__ out_m   = out + n;
    float* __restrict__ out_v   = out + 2u * n;

    if (base + 4u <= n) {
        // 128-bit non-temporal streaming loads (single-use data; keep L2 clean)
        v4f vp  = __builtin_nontemporal_load((const v4f*)(p   + base));
        v4f vp0 = __builtin_nontemporal_load((const v4f*)(p0  + base));
        v4f vg  = __builtin_nontemporal_load((const v4f*)(g   + base));
        v4f vm  = __builtin_nontemporal_load((const v4f*)(m   + base));
        v4f vv  = __builtin_nontemporal_load((const v4f*)(v   + base));
        v4f vmo = __builtin_nontemporal_load((const v4f*)(mom + base));
        v4f vnz = __builtin_nontemporal_load((const v4f*)(nz  + base));

        v4f o_mom, o_m, o_v;
#pragma unroll
        for (int k = 0; k < 4; ++k) {
            Upd r = update_one(vp[k], vp0[k], vg[k], vm[k], vv[k],
                               vmo[k], vnz[k], bc1, bc2);
            o_mom[k] = r.mom; o_m[k] = r.m; o_v[k] = r.v;
        }

        // 128-bit non-temporal streaming stores
        __builtin_nontemporal_store(o_mom, (v4f*)(out_mom + base));
        __builtin_nontemporal_store(o_m,   (v4f*)(out_m   + base));
        __builtin_nontemporal_store(o_v,   (v4f*)(out_v   + base));
    } else {
        // scalar tail (never taken for P = 2^25, kept for safety)
        for (unsigned int i = base; i < n; ++i) {
            Upd r = update_one(p[i], p0[i], g[i], m[i], v[i], mom[i], nz[i],
                               bc1, bc2);
            out_mom[i] = r.mom; out_m[i] = r.m; out_v[i] = r.v;
        }
    }
}

extern "C" void kernel_launch(void* const* d_in, const int* in_sizes, int n_in,
                              void* d_out, int out_size, void* d_ws, size_t ws_size,
                              hipStream_t stream) {
    // setup_inputs() order: p, p0, grad, m, v, mom, noise, t
    const float* p   = (const float*)d_in[0];
    const float* p0  = (const float*)d_in[1];
    const float* g   = (const float*)d_in[2];
    const float* m   = (const float*)d_in[3];
    const float* v   = (const float*)d_in[4];
    const float* mom = (const float*)d_in[5];
    const float* nz  = (const float*)d_in[6];
    const int*   t   = (const int*)d_in[7];
    float* out = (float*)d_out;

    const unsigned int n = (unsigned int)in_sizes[0];   // P = 33,554,432
    const unsigned int vec_elems = (n + 3u) / 4u;       // one float4 per thread
    const int block = 256;                              // 8 wave32 waves per block
    const int grid  = (int)((vec_elems + block - 1) / block);

    sghmc_adam_fused<<<grid, block, 0, stream>>>(p, p0, g, m, v, mom, nz, t,
                                                 out, n);
}